// _AutoformerNet_47193100648691
// MI455X (gfx1250) — compile-verified
//
#include <hip/hip_runtime.h>
#include <hip/hip_bf16.h>
#include <math.h>

typedef __attribute__((ext_vector_type(2))) float v2f;
typedef __attribute__((ext_vector_type(8))) float v8f;

#define LQ 4096
#define LMASK 4095
#define DQ 64
#define BQ 32

// -------- workspace float offsets --------
#define WS_AQ   0      // 64
#define WS_BQ   64
#define WS_AK   128
#define WS_BK   192
#define WS_AV   256
#define WS_BV   320
#define WS_CV   384    // cvec = eb + bV
#define WS_SCAL 448    // 12 scalars
#define WS_SX   464    // 32
#define WS_W0   512    // 32
#define WS_W1   544
#define WS_L0   576
#define WS_L1   608
#define WS_P    1024   // 5 * 4096
#define WS_DIAG 32768  // 32 * 256 * 32 = 262144 floats

// ---------------------------------------------------------------------------
// K0: collapse rank-1 embedding through the Q/K/V projections; LN statistics.
// aQ[n] = sum_d ew[d] qw[d,n]; bQ[n] = sum_d eb[d] qw[d,n] + qb[n]; etc.
// ---------------------------------------------------------------------------
__global__ void params_kernel(const float* __restrict__ ew, const float* __restrict__ eb,
                              const float* __restrict__ qw, const float* __restrict__ qb,
                              const float* __restrict__ kw, const float* __restrict__ kb,
                              const float* __restrict__ vw, const float* __restrict__ vb,
                              float* __restrict__ ws) {
  int n = threadIdx.x;  // 0..63
  float aq = 0.f, bq = 0.f, ak = 0.f, bk = 0.f, av = 0.f, bv = 0.f;
  for (int d = 0; d < DQ; ++d) {
    float e = ew[d], e0 = eb[d];
    aq += e * qw[d * DQ + n];  bq += e0 * qw[d * DQ + n];
    ak += e * kw[d * DQ + n];  bk += e0 * kw[d * DQ + n];
    av += e * vw[d * DQ + n];  bv += e0 * vw[d * DQ + n];
  }
  bq += qb[n]; bk += kb[n]; bv += vb[n];
  ws[WS_AQ + n] = aq; ws[WS_BQ + n] = bq;
  ws[WS_AK + n] = ak; ws[WS_BK + n] = bk;
  ws[WS_AV + n] = av; ws[WS_BV + n] = bv;
  ws[WS_CV + n] = eb[n] + bv;
  __syncthreads();
  if (n == 0) {
    float cqk = 0.f, cqb = 0.f, cbk = 0.f, cbb = 0.f;
    for (int i = 0; i < DQ; ++i) {
      cqk += ws[WS_AQ + i] * ws[WS_AK + i];
      cqb += ws[WS_AQ + i] * ws[WS_BK + i];
      cbk += ws[WS_BQ + i] * ws[WS_AK + i];
      cbb += ws[WS_BQ + i] * ws[WS_BK + i];
    }
    float mew = 0.f, mav = 0.f, mc = 0.f;
    float see = 0.f, saa = 0.f, scc = 0.f, sea = 0.f, sec = 0.f, sac = 0.f;
    for (int i = 0; i < DQ; ++i) {
      float e = ew[i], a = ws[WS_AV + i], c = ws[WS_CV + i];
      mew += e; mav += a; mc += c;
      see += e * e; saa += a * a; scc += c * c;
      sea += e * a; sec += e * c; sac += a * c;
    }
    const float inv64 = 1.0f / 64.0f;
    ws[WS_SCAL + 0] = cqk;
    ws[WS_SCAL + 1] = cqb + cbk;
    ws[WS_SCAL + 2] = cbb;
    ws[WS_SCAL + 3] = mew * inv64;  ws[WS_SCAL + 4] = mav * inv64;  ws[WS_SCAL + 5] = mc * inv64;
    ws[WS_SCAL + 6] = see * inv64;  ws[WS_SCAL + 7] = saa * inv64;  ws[WS_SCAL + 8] = scc * inv64;
    ws[WS_SCAL + 9] = sea * inv64;  ws[WS_SCAL + 10] = sec * inv64; ws[WS_SCAL + 11] = sac * inv64;
  }
}

// ---------------------------------------------------------------------------
// K1: Sx[b] = sum_t x[b,t]
// ---------------------------------------------------------------------------
__global__ void sumx_kernel(const float* __restrict__ x, float* __restrict__ ws) {
  __shared__ float red[256];
  int b = blockIdx.x, t = threadIdx.x;
  float s = 0.f;
  for (int i = t; i < LQ; i += 256) s += x[b * LQ + i];
  red[t] = s; __syncthreads();
  for (int st = 128; st > 0; st >>= 1) {
    if (t < st) red[t] += red[t + st];
    __syncthreads();
  }
  if (t == 0) ws[WS_SX + b] = red[0];
}

// ---------------------------------------------------------------------------
// K2: scalar circular autocorrelation of x via v_wmma_f32_16x16x4_f32.
// Wave handles (b, Delta=16*delta): A[i,k]=x[tb+16k+i], B[k,j]=x[tb-Delta+16k+j]
// so every product in the tile has lag Delta+(i-j); accumulate over all tb,
// then reduce the tile to 31 diagonal sums in LDS (deterministic, no atomics).
// Block = 8 waves sharing x[b] staged in LDS.
// ---------------------------------------------------------------------------
__global__ void autocorr_wmma(const float* __restrict__ x, float* __restrict__ ws) {
  __shared__ float xs[LQ];
  __shared__ float gt[8 * 256];
  int bt = blockIdx.y;
  for (int i = threadIdx.x; i < LQ; i += 256) xs[i] = x[bt * LQ + i];
  __syncthreads();

  int wave = threadIdx.x >> 5;
  int lane = threadIdx.x & 31;
  int delta = blockIdx.x * 8 + wave;     // 0..255
  int Dlag = delta * 16;
  int hi = (lane >= 16) ? 1 : 0;
  int m = lane & 15;
  int off0 = m + (hi ? 32 : 0);          // K = 0 or 2 slot
  int off1 = off0 + 16;                  // K = 1 or 3 slot

  v8f acc = {};
  for (int tb = 0; tb < LQ; tb += 64) {
    v2f a, bb;
    a.x = xs[tb + off0];
    a.y = xs[tb + off1];
    bb.x = xs[(tb + off0 - Dlag) & LMASK];
    bb.y = xs[(tb + off1 - Dlag) & LMASK];
    acc = __builtin_amdgcn_wmma_f32_16x16x4_f32(false, a, false, bb,
                                                (short)0, acc, false, false);
  }

  // lane holds G[i=r+8*hi][j=m] in acc[r]; spill tile to LDS
  int i0 = hi * 8;
#pragma unroll
  for (int r = 0; r < 8; ++r) gt[wave * 256 + (i0 + r) * 16 + m] = acc[r];
  __syncthreads();

  // diagonal sums: cp = i - j in [-15,15] -> lane = cp + 15
  if (lane < 31) {
    int cp = lane - 15;
    float s = 0.f;
#pragma unroll
    for (int i = 0; i < 16; ++i) {
      int j = i - cp;
      if (j >= 0 && j < 16) s += gt[wave * 256 + i * 16 + j];
    }
    ws[WS_DIAG + (bt * 256 + delta) * 32 + lane] = s;
  }
}

// ---------------------------------------------------------------------------
// K3: per-b top-2 of mean_corr (built on the fly from diagonal sums) + softmax
// mean_corr[b,l] = (cQK*r_xx + cS*Sx[b] + L*cbb)/64 ; lag l = 16*d1 + c comes
// from DIAG[d1][c+15] and (c>0) DIAG[d1+1][c-1].
// ---------------------------------------------------------------------------
__global__ void top2_kernel(float* __restrict__ ws) {
  int b = blockIdx.x, t = threadIdx.x;
  float cQK = ws[WS_SCAL + 0], cS = ws[WS_SCAL + 1], cbb = ws[WS_SCAL + 2];
  float base = cS * ws[WS_SX + b] + (float)LQ * cbb;
  const float* DG = ws + WS_DIAG + b * 256 * 32;

  float v0 = -3.4e38f, v1 = -3.4e38f; int i0 = 1 << 30, i1 = 1 << 30;
  for (int l = t; l < LQ; l += 256) {
    int cc = l & 15;
    int d1 = l >> 4;
    float r = DG[d1 * 32 + cc + 15];
    if (cc) r += DG[((d1 + 1) & 255) * 32 + cc - 1];
    float v = (cQK * r + base) * (1.0f / 64.0f);
    if (v > v0 || (v == v0 && l < i0)) { v1 = v0; i1 = i0; v0 = v; i0 = l; }
    else if (v > v1 || (v == v1 && l < i1)) { v1 = v; i1 = l; }
  }
  __shared__ float sv0[256], sv1[256];
  __shared__ int   si0[256], si1[256];
  sv0[t] = v0; sv1[t] = v1; si0[t] = i0; si1[t] = i1;
  __syncthreads();
  for (int st = 128; st > 0; st >>= 1) {
    if (t < st) {
      float a0 = sv0[t], a1 = sv1[t], c0 = sv0[t + st], c1 = sv1[t + st];
      int a0i = si0[t], a1i = si1[t], c0i = si0[t + st], c1i = si1[t + st];
      bool aFirst = (a0 > c0) || (a0 == c0 && a0i < c0i);
      if (aFirst) {
        bool aSec = (a1 > c0) || (a1 == c0 && a1i < c0i);
        sv0[t] = a0; si0[t] = a0i;
        sv1[t] = aSec ? a1 : c0; si1[t] = aSec ? a1i : c0i;
      } else {
        bool cSec = (c1 > a0) || (c1 == a0 && c1i < a0i);
        sv0[t] = c0; si0[t] = c0i;
        sv1[t] = cSec ? c1 : a0; si1[t] = cSec ? c1i : a0i;
      }
    }
    __syncthreads();
  }
  if (t == 0) {
    float e1 = expf(sv1[0] - sv0[0]);      // v0 >= v1 (sorted)
    float w0 = 1.0f / (1.0f + e1);
    ws[WS_W0 + b] = w0;
    ws[WS_W1 + b] = e1 * w0;
    ws[WS_L0 + b] = (float)si0[0];
    ws[WS_L1 + b] = (float)si1[0];
  }
}

// ---------------------------------------------------------------------------
// K4: P1..P5[l] — moving-average is self-adjoint, so fold it into fc_w once:
// g[l,d] = fc_w[l,d] - (1/25) * sum_{l' in [l-12,l+12] ∩ [0,L)} fc_w[l',d]
// ---------------------------------------------------------------------------
__global__ void pvec_kernel(const float* __restrict__ fcw, const float* __restrict__ ew,
                            const float* __restrict__ lng, const float* __restrict__ lnb,
                            float* __restrict__ ws) {
  int l = blockIdx.x * 256 + threadIdx.x;
  if (l >= LQ) return;
  int lo = l - 12; if (lo < 0) lo = 0;
  int hiw = l + 12; if (hiw > LQ - 1) hiw = LQ - 1;
  float p1 = 0.f, p2 = 0.f, p3 = 0.f, p4 = 0.f, p5 = 0.f;
  for (int d = 0; d < DQ; ++d) {
    float s = 0.f;
    for (int lp = lo; lp <= hiw; ++lp) s += fcw[lp * DQ + d];
    float g = fcw[l * DQ + d] - s * (1.0f / 25.0f);
    float lg = lng[d];
    p1 += g * ew[d] * lg;
    p2 += g * ws[WS_AV + d] * lg;
    p3 += g * ws[WS_CV + d] * lg;
    p4 += g * lg;
    p5 += g * lnb[d];
  }
  ws[WS_P + l]            = p1;
  ws[WS_P + LQ + l]       = p2;
  ws[WS_P + 2 * LQ + l]   = p3;
  ws[WS_P + 3 * LQ + l]   = p4;
  ws[WS_P + 4 * LQ + l]   = p5;
}

// ---------------------------------------------------------------------------
// K5: out[b] = fc_b + sum_l [ inv*(u*P1 + v*P2 + P3 - mu*P4) + P5 ]
// u = x[b,l]; v = w0*x[b,(l-lag0)%L] + w1*x[b,(l-lag1)%L]
// mu / var from closed-form second moments over d.
// ---------------------------------------------------------------------------
__global__ void final_kernel(const float* __restrict__ x, const float* __restrict__ ws,
                             const float* __restrict__ fcb, float* __restrict__ out) {
  int b = blockIdx.x, t = threadIdx.x;
  float w0 = ws[WS_W0 + b], w1 = ws[WS_W1 + b];
  int lag0 = (int)ws[WS_L0 + b], lag1 = (int)ws[WS_L1 + b];
  float mew = ws[WS_SCAL + 3], mav = ws[WS_SCAL + 4], mc = ws[WS_SCAL + 5];
  float see = ws[WS_SCAL + 6], saa = ws[WS_SCAL + 7], scc = ws[WS_SCAL + 8];
  float sea = ws[WS_SCAL + 9], sec = ws[WS_SCAL + 10], sac = ws[WS_SCAL + 11];
  const float* xb = x + b * LQ;
  const float* P = ws + WS_P;
  float acc = 0.f;
  for (int l = t; l < LQ; l += 256) {
    float u = xb[l];
    float v = w0 * xb[(l - lag0) & LMASK] + w1 * xb[(l - lag1) & LMASK];
    float mu = u * mew + v * mav + mc;
    float ez2 = u * u * see + v * v * saa + scc
              + 2.f * (u * v * sea + u * sec + v * sac);
    float var = ez2 - mu * mu;
    float inv = rsqrtf(var + 1e-5f);
    acc += inv * (u * P[l] + v * P[LQ + l] + P[2 * LQ + l] - mu * P[3 * LQ + l])
         + P[4 * LQ + l];
  }
  __shared__ float red[256];
  red[t] = acc; __syncthreads();
  for (int st = 128; st > 0; st >>= 1) {
    if (t < st) red[t] += red[t + st];
    __syncthreads();
  }
  if (t == 0) out[b] = red[0] + fcb[0];
}

// ---------------------------------------------------------------------------
extern "C" void kernel_launch(void* const* d_in, const int* in_sizes, int n_in,
                              void* d_out, int out_size, void* d_ws, size_t ws_size,
                              hipStream_t stream) {
  const float* x   = (const float*)d_in[0];
  const float* ew  = (const float*)d_in[1];
  const float* eb  = (const float*)d_in[2];
  const float* qw  = (const float*)d_in[3];
  const float* qb  = (const float*)d_in[4];
  const float* kw  = (const float*)d_in[5];
  const float* kb  = (const float*)d_in[6];
  const float* vw  = (const float*)d_in[7];
  const float* vb  = (const float*)d_in[8];
  const float* lng = (const float*)d_in[9];
  const float* lnb = (const float*)d_in[10];
  const float* fcw = (const float*)d_in[11];
  const float* fcb = (const float*)d_in[12];
  float* out = (float*)d_out;
  float* ws  = (float*)d_ws;

  params_kernel<<<1, 64, 0, stream>>>(ew, eb, qw, qb, kw, kb, vw, vb, ws);
  sumx_kernel<<<BQ, 256, 0, stream>>>(x, ws);
  dim3 g1(32, BQ);                      // 32 delta-groups x 32 batches, 8 waves/block
  autocorr_wmma<<<g1, 256, 0, stream>>>(x, ws);
  top2_kernel<<<BQ, 256, 0, stream>>>(ws);
  pvec_kernel<<<LQ / 256, 256, 0, stream>>>(fcw, ew, lng, lnb, ws);
  final_kernel<<<BQ, 256, 0, stream>>>(x, ws, fcb, out);
}